// Composition_58360015618223
// MI455X (gfx1250) — compile-verified
//
#include <hip/hip_runtime.h>
#include <hip/hip_bf16.h>
#include <math.h>

typedef __attribute__((ext_vector_type(2))) float v2f;
typedef __attribute__((ext_vector_type(8))) float v8f;

#define SPH_EPS 1e-12f

// ---------------------------------------------------------------------------
// Kernel 0: standardize y, advect pos/vel. Packs |x|^2 into pos4.w and x.v
// into vel4.w (the WMMA K=3 pad is supplied as a literal 0 when building
// operands, so .w is free to carry the per-particle scalars).
// ---------------------------------------------------------------------------
__global__ __launch_bounds__(256) void sph_prep(
    const float* __restrict__ y,
    const float* __restrict__ mid_pos,
    const float* __restrict__ mid_vel,
    const float* __restrict__ y_mean,
    const float* __restrict__ y_std,
    const float* __restrict__ dtp,
    const int* __restrict__ nbp,
    float4* __restrict__ pos4,
    float4* __restrict__ vel4,
    int N)
{
    int i = blockIdx.x * blockDim.x + threadIdx.x;
    if (i >= N) return;
    const int nb = *nbp;
    const float dt = *dtp;

    float y0 = y[3 * i + 0] * y_std[0] + y_mean[0];
    float y1 = y[3 * i + 1] * y_std[1] + y_mean[1];
    float y2 = y[3 * i + 2] * y_std[2] + y_mean[2];
    const bool fluid = (i >= nb);

    float px = mid_pos[3 * i + 0] + (fluid ? y0 : 0.0f);
    float py = mid_pos[3 * i + 1] + (fluid ? y1 : 0.0f);
    float pz = mid_pos[3 * i + 2] + (fluid ? y2 : 0.0f);

    float vx = mid_vel[3 * i + 0] + (fluid ? y0 / dt : 0.0f);
    float vy = mid_vel[3 * i + 1] + (fluid ? y1 / dt : 0.0f);
    float vz = mid_vel[3 * i + 2] + (fluid ? y2 / dt : 0.0f);

    float sqv = px * px + py * py + pz * pz;
    float dgv = px * vx + py * vy + pz * vz;
    pos4[i] = make_float4(px, py, pz, sqv);
    vel4[i] = make_float4(vx, vy, vz, dgv);
}

// ---------------------------------------------------------------------------
// Kernel 1: all-pairs tiles. One block per 16-row i-tile; 8 waves split the
// j-tiles. Per 16x16 tile: 3x V_WMMA_F32_16X16X4_F32 compute
//   G  = x_i . x_j,  Dm = x_i . v_j,  Dn = v_i . x_j
// Branchless cubic-spline epilogue:
//   a = max(1-q,0), b = max(0.5-q,0)
//   W_core  = 2a^3 - 8b^3          (== reference on both pieces, 0 for q>1)
//   dW_core = 24b^2 - 6a^2         (sigma, sigma/h applied in finalize)
//
// f32 16x4 A layout (ISA 7.12.2): lanes 0-15 hold M=lane, VGPR0=K0, VGPR1=K1;
// lanes 16-31 hold M=lane-16, VGPR0=K2, VGPR1=K3 (pad = literal 0).
// B mirrors with N=lane%16.  C/D: VGPR v -> row m = v + 8*(lane/16), col n.
// ---------------------------------------------------------------------------
__global__ __launch_bounds__(256) void sph_pairs(
    const float4* __restrict__ pos4,
    const float4* __restrict__ vel4,
    const float* __restrict__ hp,
    float* __restrict__ wsum_g,
    float* __restrict__ dsum_g,
    int N)
{
    const int lane   = threadIdx.x & 31;
    const int wave   = threadIdx.x >> 5;
    const int nwaves = blockDim.x >> 5;
    const int it     = blockIdx.x;
    const int ntiles = N >> 4;

    const float h     = *hp;
    const float inv_h = 1.0f / h;

    const int  col = lane & 15;
    const bool hi  = (lane >= 16);
    const int  mbase = hi ? 8 : 0;

    // A operands for this i-tile (constant across the j loop)
    const float4 pi = pos4[(it << 4) + col];
    const float4 vi = vel4[(it << 4) + col];
    v2f A;  A.x  = hi ? pi.z : pi.x;  A.y  = hi ? 0.0f : pi.y;
    v2f VA; VA.x = hi ? vi.z : vi.x;  VA.y = hi ? 0.0f : vi.y;

    // Row-wise |x_i|^2 and x_i.v_i, gathered once from the packed .w slots.
    const float* pos_f = (const float*)pos4;
    const float* vel_f = (const float*)vel4;
    float sqi[8], dgi[8];
#pragma unroll
    for (int v = 0; v < 8; ++v) {
        int idx = (it << 4) + mbase + v;
        sqi[v] = pos_f[4 * idx + 3];
        dgi[v] = vel_f[4 * idx + 3];
    }

    float wsum[8], dsum[8];
#pragma unroll
    for (int v = 0; v < 8; ++v) { wsum[v] = 0.0f; dsum[v] = 0.0f; }

    for (int jt = wave; jt < ntiles; jt += nwaves) {
        const int j = (jt << 4) + col;
        const float4 pj = pos4[j];
        const float4 vj = vel4[j];
        const float sqj = pj.w;
        const float dgj = vj.w;

        v2f B;  B.x  = hi ? pj.z : pj.x;  B.y  = hi ? 0.0f : pj.y;
        v2f VB; VB.x = hi ? vj.z : vj.x;  VB.y = hi ? 0.0f : vj.y;

        v8f zero = {};
        v8f G  = __builtin_amdgcn_wmma_f32_16x16x4_f32(
            false, A,  false, B,  (short)0, zero, false, false);
        v8f Dm = __builtin_amdgcn_wmma_f32_16x16x4_f32(
            false, A,  false, VB, (short)0, zero, false, false);
        v8f Dn = __builtin_amdgcn_wmma_f32_16x16x4_f32(
            false, VA, false, B,  (short)0, zero, false, false);

#pragma unroll
        for (int v = 0; v < 8; ++v) {
            float d2  = fmaxf(fmaf(-2.0f, G[v], sqi[v] + sqj), 0.0f);
            float dmx = fmaxf(d2, SPH_EPS);
            float irr = __builtin_amdgcn_rsqf(dmx);   // v_rsq_f32
            float q   = dmx * irr * inv_h;            // r/h
            float a   = fmaxf(1.0f - q, 0.0f);
            float b   = fmaxf(0.5f - q, 0.0f);
            float a2  = a * a;
            float b2  = b * b;
            wsum[v]  += fmaf(2.0f * a, a2, -8.0f * b * b2);
            float inv_r = (d2 > SPH_EPS) ? irr : 0.0f;
            float dw    = fmaf(24.0f, b2, -6.0f * a2);
            float dot   = (Dm[v] + Dn[v]) - (dgj + dgi[v]);
            dsum[v]     = fmaf(dw * inv_r, dot, dsum[v]);
        }
    }

    // Reduce across the 16 lanes of each half-wave (columns of the tile);
    // rows m=0..7 live in lanes 0-15, rows m=8..15 in lanes 16-31.
#pragma unroll
    for (int v = 0; v < 8; ++v) {
#pragma unroll
        for (int off = 8; off >= 1; off >>= 1) {
            wsum[v] += __shfl_xor(wsum[v], off, 32);
            dsum[v] += __shfl_xor(dsum[v], off, 32);
        }
    }

    __shared__ float lds_w[8][16];
    __shared__ float lds_d[8][16];
    if (col == 0) {  // lane 0 holds rows 0..7, lane 16 holds rows 8..15
#pragma unroll
        for (int v = 0; v < 8; ++v) {
            lds_w[wave][mbase + v] = wsum[v];
            lds_d[wave][mbase + v] = dsum[v];
        }
    }
    __syncthreads();

    if (threadIdx.x < 16) {
        float tw = 0.0f, td = 0.0f;
#pragma unroll
        for (int w = 0; w < 8; ++w) {
            tw += lds_w[w][threadIdx.x];
            td += lds_d[w][threadIdx.x];
        }
        wsum_g[(it << 4) + threadIdx.x] = tw;
        dsum_g[(it << 4) + threadIdx.x] = td;
    }
}

// ---------------------------------------------------------------------------
// Kernel 2: final losses. One block, fixed-order tree reduction.
// rho/rho_0 - 1 = vol*sigma*Wsum - 1 (rho_0 cancels); div = vol*(sigma/h)*Dsum
// ---------------------------------------------------------------------------
__global__ __launch_bounds__(256) void sph_finalize(
    const float* __restrict__ pred,
    const float* __restrict__ y,
    const float* __restrict__ wsum_g,
    const float* __restrict__ dsum_g,
    const float* __restrict__ hp,
    const float* __restrict__ volp,
    float* __restrict__ out,
    int N)
{
    const int tid = threadIdx.x;
    const float h    = *hp;
    const float volv = *volp;
    const float sigma = 8.0f / (3.14159265358979323846f * h * h * h);
    const float c2 = volv * sigma;
    const float c3 = volv * sigma / h;

    float s1 = 0.0f, s2 = 0.0f, s3 = 0.0f;
    for (int k = tid; k < 3 * N; k += 256) {
        float d = y[k] - pred[k];
        s1 = fmaf(d, d, s1);
    }
    for (int i = tid; i < N; i += 256) {
        s2 += fabsf(c2 * wsum_g[i] - 1.0f);
        s3 += fabsf(c3 * dsum_g[i]);
    }

    __shared__ float r1[256], r2[256], r3[256];
    r1[tid] = s1; r2[tid] = s2; r3[tid] = s3;
    __syncthreads();
    for (int s = 128; s > 0; s >>= 1) {
        if (tid < s) {
            r1[tid] += r1[tid + s];
            r2[tid] += r2[tid + s];
            r3[tid] += r3[tid + s];
        }
        __syncthreads();
    }
    if (tid == 0) {
        float invN = 1.0f / (float)N;
        out[0] = r1[0] * invN + 0.1f * (r2[0] * invN) + 0.1f * (r3[0] * invN);
    }
}

// ---------------------------------------------------------------------------
extern "C" void kernel_launch(void* const* d_in, const int* in_sizes, int n_in,
                              void* d_out, int out_size, void* d_ws, size_t ws_size,
                              hipStream_t stream) {
    const float* pred    = (const float*)d_in[0];
    const float* y       = (const float*)d_in[1];
    const float* mid_pos = (const float*)d_in[2];
    const float* mid_vel = (const float*)d_in[3];
    const float* y_mean  = (const float*)d_in[4];
    const float* y_std   = (const float*)d_in[5];
    const float* hp      = (const float*)d_in[6];
    const float* volp    = (const float*)d_in[7];
    // d_in[8] = rho_0 (cancels in loss2, unused)
    const float* dtp     = (const float*)d_in[9];
    const int*   nbp     = (const int*)d_in[10];

    const int N = in_sizes[2] / 3;   // mid_pos is (N,3)
    const int ntiles = N / 16;

    float* ws = (float*)d_ws;
    float4* pos4 = (float4*)(ws);                    // 4N floats
    float4* vel4 = (float4*)(ws + (size_t)4 * N);    // 4N floats
    float*  wsum = ws + (size_t)8 * N;               // N
    float*  dsum = ws + (size_t)9 * N;               // N

    sph_prep<<<(N + 255) / 256, 256, 0, stream>>>(
        y, mid_pos, mid_vel, y_mean, y_std, dtp, nbp,
        pos4, vel4, N);

    sph_pairs<<<ntiles, 256, 0, stream>>>(
        pos4, vel4, hp, wsum, dsum, N);

    sph_finalize<<<1, 256, 0, stream>>>(
        pred, y, wsum, dsum, hp, volp, (float*)d_out, N);
}